// MMAELocal_42563125903682
// MI455X (gfx1250) — compile-verified
//
#include <hip/hip_runtime.h>
#include <hip/hip_bf16.h>
#include <float.h>

#define B_N  4096
#define D_X  1024
#define D_Z  128
#define KMAX 12
#define EPSF 1e-8f

typedef __attribute__((ext_vector_type(2))) float v2f;
typedef __attribute__((ext_vector_type(8))) float v8f;

// D = A(16x4 f32) * B(4x16 f32) + C, exact fp32 matrix op on the WMMA pipe.
static __device__ __forceinline__ v8f wmma_f32(v2f a, v2f b, v8f c) {
    return __builtin_amdgcn_wmma_f32_16x16x4_f32(
        /*neg_a=*/false, a, /*neg_b=*/false, b,
        /*c_mod=*/(short)0, c, /*reuse_a=*/false, /*reuse_b=*/false);
}

// ---------------------------------------------------------------- init
__global__ void mmae_init_kernel(float* maxz, float* maxr, float* errs, float* out) {
    if (threadIdx.x == 0) { maxz[0] = 0.f; maxr[0] = 0.f; errs[0] = 0.f; out[0] = 0.f; }
}

// ---------------------------------------------------------------- row squared norms
__global__ void mmae_sqnorm_kernel(const float* __restrict__ X, int D, float* __restrict__ sq) {
    int i = blockIdx.x;
    const float* row = X + (size_t)i * D;
    float s = 0.f;
    for (int d = threadIdx.x; d < D; d += blockDim.x) { float v = row[d]; s += v * v; }
    __shared__ float sm[256];
    sm[threadIdx.x] = s; __syncthreads();
    for (int o = 128; o > 0; o >>= 1) {
        if ((int)threadIdx.x < o) sm[threadIdx.x] += sm[threadIdx.x + o];
        __syncthreads();
    }
    if (threadIdx.x == 0) sq[i] = sm[0];
}

// ---------------------------------------------------------------- G = X * X^T (fp32 WMMA)
// One wave computes a 16x64 strip: A fragment reused across 4 N-tiles
// (1.25 memory ops per WMMA instead of 2).
__global__ void mmae_gemm_xxT_kernel(const float* __restrict__ X, float* __restrict__ G) {
    const int lane = threadIdx.x;          // 0..31
    const int half = lane >> 4;            // 0: K={0,1}, 1: K={2,3}
    const int l    = lane & 15;
    const int tj = blockIdx.x;             // 64-wide column strip
    const int ti = blockIdx.y;             // 16-row strip
    const float* arow = X + (size_t)(ti * 16 + l) * D_X;          // A rows: M = l
    const float* b0 = X + (size_t)(tj * 64 +  0 + l) * D_X;       // B cols (B = X^T tile)
    const float* b1 = X + (size_t)(tj * 64 + 16 + l) * D_X;
    const float* b2 = X + (size_t)(tj * 64 + 32 + l) * D_X;
    const float* b3 = X + (size_t)(tj * 64 + 48 + l) * D_X;
    v8f c0 = {}, c1 = {}, c2 = {}, c3 = {};
    for (int kk = 0; kk < D_X; kk += 4) {
        const int o = kk + 2 * half;
        v2f a; a.x = arow[o]; a.y = arow[o + 1];
        v2f v;
        v.x = b0[o]; v.y = b0[o + 1]; c0 = wmma_f32(a, v, c0);
        v.x = b1[o]; v.y = b1[o + 1]; c1 = wmma_f32(a, v, c1);
        v.x = b2[o]; v.y = b2[o + 1]; c2 = wmma_f32(a, v, c2);
        v.x = b3[o]; v.y = b3[o + 1]; c3 = wmma_f32(a, v, c3);
    }
    // C/D layout: VGPR r -> row (r + 8*half), col = lane&15 within each 16-tile
    for (int r = 0; r < 8; ++r) {
        int gi = ti * 16 + r + 8 * half;
        float* grow = G + (size_t)gi * B_N + tj * 64 + l;
        grow[0] = c0[r]; grow[16] = c1[r]; grow[32] = c2[r]; grow[48] = c3[r];
    }
}

// ---------------------------------------------------------------- top-k nearest neighbors
__global__ void mmae_topk_kernel(const float* __restrict__ G, const float* __restrict__ sq,
                                 const int* __restrict__ kptr, int* __restrict__ knn) {
    const int i = blockIdx.x;
    const int tid = threadIdx.x;
    int k = kptr[0]; if (k > KMAX) k = KMAX; if (k < 1) k = 1;
    __shared__ float row[B_N];
    __shared__ float rv[256];
    __shared__ int   ri[256];
    const float sqi = sq[i];
    for (int j = tid; j < B_N; j += 256)
        row[j] = (j == i) ? FLT_MAX : (sqi + sq[j] - 2.f * G[(size_t)i * B_N + j]);
    __syncthreads();
    for (int t = 0; t < k; ++t) {
        float best = FLT_MAX; int bj = 0;
        for (int j = tid; j < B_N; j += 256) {
            float v = row[j];
            if (v < best) { best = v; bj = j; }
        }
        rv[tid] = best; ri[tid] = bj; __syncthreads();
        for (int o = 128; o > 0; o >>= 1) {
            if (tid < o && rv[tid + o] < rv[tid]) { rv[tid] = rv[tid + o]; ri[tid] = ri[tid + o]; }
            __syncthreads();
        }
        if (tid == 0) { knn[i * KMAX + t] = ri[0]; row[ri[0]] = FLT_MAX; }
        __syncthreads();
    }
}

// ---------------------------------------------------------------- local curvature (SVD via k x k Jacobi)
__global__ void mmae_curv_kernel(const float* __restrict__ X, const int* __restrict__ knn,
                                 const int* __restrict__ kptr, float* __restrict__ curv) {
    const int i = blockIdx.x;
    const int tid = threadIdx.x;
    int k = kptr[0]; if (k > KMAX) k = KMAX; if (k < 1) k = 1;
    __shared__ float nb[KMAX * D_X];                 // 48 KB
    __shared__ float cov[KMAX * (KMAX + 1) / 2];
    __shared__ float M[KMAX][KMAX];
    __shared__ int   nidx[KMAX];
    if (tid < k) nidx[tid] = knn[i * KMAX + tid];
    __syncthreads();
    for (int a = 0; a < k; ++a) {
        const float* src = X + (size_t)nidx[a] * D_X;
        for (int d = tid; d < D_X; d += 256) nb[a * D_X + d] = src[d];
    }
    __syncthreads();
    // center across the k neighbors
    for (int d = tid; d < D_X; d += 256) {
        float m = 0.f;
        for (int a = 0; a < k; ++a) m += nb[a * D_X + d];
        m /= (float)k;
        for (int a = 0; a < k; ++a) nb[a * D_X + d] -= m;
    }
    __syncthreads();
    // Gram: one warp per (a,b) pair
    const int npair = k * (k + 1) / 2;
    const int wid = tid >> 5, lane = tid & 31;
    for (int p = wid; p < npair; p += 8) {
        int a = 0, pp = p, rowlen = k;
        while (pp >= rowlen) { pp -= rowlen; ++a; --rowlen; }
        int b = a + pp;
        float s = 0.f;
        for (int d = lane; d < D_X; d += 32) s += nb[a * D_X + d] * nb[b * D_X + d];
        for (int o = 16; o > 0; o >>= 1) s += __shfl_xor(s, o);
        if (lane == 0) cov[p] = s;
    }
    __syncthreads();
    if (tid == 0) {
        int p = 0;
        for (int a = 0; a < k; ++a)
            for (int b = a; b < k; ++b) { M[a][b] = cov[p]; M[b][a] = cov[p]; ++p; }
        // cyclic Jacobi eigensolve
        for (int sweep = 0; sweep < 30; ++sweep) {
            for (int a = 0; a < k - 1; ++a)
                for (int b = a + 1; b < k; ++b) {
                    float apq = M[a][b];
                    if (fabsf(apq) < 1e-12f) continue;
                    float app = M[a][a], aqq = M[b][b];
                    float tau = (aqq - app) / (2.f * apq);
                    float t = ((tau >= 0.f) ? 1.f : -1.f) / (fabsf(tau) + sqrtf(1.f + tau * tau));
                    float cc = rsqrtf(1.f + t * t);
                    float ss = t * cc;
                    for (int m = 0; m < k; ++m) {
                        float mp = M[m][a], mq = M[m][b];
                        M[m][a] = cc * mp - ss * mq;
                        M[m][b] = ss * mp + cc * mq;
                    }
                    for (int m = 0; m < k; ++m) {
                        float pm = M[a][m], qm = M[b][m];
                        M[a][m] = cc * pm - ss * qm;
                        M[b][m] = ss * pm + cc * qm;
                    }
                }
        }
        float smax = 0.f, ssum = 0.f;
        for (int a = 0; a < k; ++a) {
            float lam = M[a][a];
            float sv = (lam > 0.f) ? sqrtf(lam) : 0.f;
            ssum += sv;
            if (sv > smax) smax = sv;
        }
        curv[i] = 1.f - smax / (ssum + EPSF);
    }
}

// ---------------------------------------------------------------- pass 1: max of z/ref distance matrices
// One wave computes a 16x64 strip of both z and ref distance tiles (A reuse x4 each).
__global__ void mmae_zr_max_kernel(const float* __restrict__ Z, const float* __restrict__ R,
                                   const float* __restrict__ sqz, const float* __restrict__ sqr,
                                   float* maxz, float* maxr) {
    const int lane = threadIdx.x, half = lane >> 4, l = lane & 15;
    const int tj = blockIdx.x, ti = blockIdx.y;
    const float* za  = Z + (size_t)(ti * 16 + l) * D_Z;
    const float* ra  = R + (size_t)(ti * 16 + l) * D_Z;
    const float* zb0 = Z + (size_t)(tj * 64 +  0 + l) * D_Z;
    const float* zb1 = Z + (size_t)(tj * 64 + 16 + l) * D_Z;
    const float* zb2 = Z + (size_t)(tj * 64 + 32 + l) * D_Z;
    const float* zb3 = Z + (size_t)(tj * 64 + 48 + l) * D_Z;
    const float* rb0 = R + (size_t)(tj * 64 +  0 + l) * D_Z;
    const float* rb1 = R + (size_t)(tj * 64 + 16 + l) * D_Z;
    const float* rb2 = R + (size_t)(tj * 64 + 32 + l) * D_Z;
    const float* rb3 = R + (size_t)(tj * 64 + 48 + l) * D_Z;
    v8f cz0 = {}, cz1 = {}, cz2 = {}, cz3 = {};
    v8f cr0 = {}, cr1 = {}, cr2 = {}, cr3 = {};
    for (int kk = 0; kk < D_Z; kk += 4) {
        const int o = kk + 2 * half;
        v2f a, v;
        a.x = za[o]; a.y = za[o + 1];
        v.x = zb0[o]; v.y = zb0[o + 1]; cz0 = wmma_f32(a, v, cz0);
        v.x = zb1[o]; v.y = zb1[o + 1]; cz1 = wmma_f32(a, v, cz1);
        v.x = zb2[o]; v.y = zb2[o + 1]; cz2 = wmma_f32(a, v, cz2);
        v.x = zb3[o]; v.y = zb3[o + 1]; cz3 = wmma_f32(a, v, cz3);
        a.x = ra[o]; a.y = ra[o + 1];
        v.x = rb0[o]; v.y = rb0[o + 1]; cr0 = wmma_f32(a, v, cr0);
        v.x = rb1[o]; v.y = rb1[o + 1]; cr1 = wmma_f32(a, v, cr1);
        v.x = rb2[o]; v.y = rb2[o + 1]; cr2 = wmma_f32(a, v, cr2);
        v.x = rb3[o]; v.y = rb3[o + 1]; cr3 = wmma_f32(a, v, cr3);
    }
    float mz = 0.f, mr = 0.f;
    for (int nt = 0; nt < 4; ++nt) {
        const v8f* cz = (nt == 0) ? &cz0 : (nt == 1) ? &cz1 : (nt == 2) ? &cz2 : &cz3;
        const v8f* cr = (nt == 0) ? &cr0 : (nt == 1) ? &cr1 : (nt == 2) ? &cr2 : &cr3;
        for (int r = 0; r < 8; ++r) {
            int gi = ti * 16 + r + 8 * half;
            int gj = tj * 64 + nt * 16 + l;
            float d2 = sqz[gi] + sqz[gj] - 2.f * (*cz)[r];
            float dd = (d2 > 0.f) ? sqrtf(d2) : 0.f;
            if (dd > mz) mz = dd;
            d2 = sqr[gi] + sqr[gj] - 2.f * (*cr)[r];
            dd = (d2 > 0.f) ? sqrtf(d2) : 0.f;
            if (dd > mr) mr = dd;
        }
    }
    for (int o = 16; o > 0; o >>= 1) {
        mz = fmaxf(mz, __shfl_xor(mz, o));
        mr = fmaxf(mr, __shfl_xor(mr, o));
    }
    if (lane == 0) {
        atomicMax((unsigned int*)maxz, __float_as_uint(mz));   // all values >= 0
        atomicMax((unsigned int*)maxr, __float_as_uint(mr));
    }
}

// ---------------------------------------------------------------- pass 2: weighted masked error sum
__global__ void mmae_err_kernel(const float* __restrict__ Z, const float* __restrict__ R,
                                const float* __restrict__ sqz, const float* __restrict__ sqr,
                                const float* __restrict__ curv,
                                const float* __restrict__ maxz, const float* __restrict__ maxr,
                                float* __restrict__ errs) {
    const int lane = threadIdx.x, half = lane >> 4, l = lane & 15;
    const int tj = blockIdx.x, ti = blockIdx.y;
    if ((tj * 64 + 64) <= (ti * 16)) return;    // whole strip strictly below diagonal
    const float* za  = Z + (size_t)(ti * 16 + l) * D_Z;
    const float* ra  = R + (size_t)(ti * 16 + l) * D_Z;
    const float* zb0 = Z + (size_t)(tj * 64 +  0 + l) * D_Z;
    const float* zb1 = Z + (size_t)(tj * 64 + 16 + l) * D_Z;
    const float* zb2 = Z + (size_t)(tj * 64 + 32 + l) * D_Z;
    const float* zb3 = Z + (size_t)(tj * 64 + 48 + l) * D_Z;
    const float* rb0 = R + (size_t)(tj * 64 +  0 + l) * D_Z;
    const float* rb1 = R + (size_t)(tj * 64 + 16 + l) * D_Z;
    const float* rb2 = R + (size_t)(tj * 64 + 32 + l) * D_Z;
    const float* rb3 = R + (size_t)(tj * 64 + 48 + l) * D_Z;
    v8f cz0 = {}, cz1 = {}, cz2 = {}, cz3 = {};
    v8f cr0 = {}, cr1 = {}, cr2 = {}, cr3 = {};
    for (int kk = 0; kk < D_Z; kk += 4) {
        const int o = kk + 2 * half;
        v2f a, v;
        a.x = za[o]; a.y = za[o + 1];
        v.x = zb0[o]; v.y = zb0[o + 1]; cz0 = wmma_f32(a, v, cz0);
        v.x = zb1[o]; v.y = zb1[o + 1]; cz1 = wmma_f32(a, v, cz1);
        v.x = zb2[o]; v.y = zb2[o + 1]; cz2 = wmma_f32(a, v, cz2);
        v.x = zb3[o]; v.y = zb3[o + 1]; cz3 = wmma_f32(a, v, cz3);
        a.x = ra[o]; a.y = ra[o + 1];
        v.x = rb0[o]; v.y = rb0[o + 1]; cr0 = wmma_f32(a, v, cr0);
        v.x = rb1[o]; v.y = rb1[o + 1]; cr1 = wmma_f32(a, v, cr1);
        v.x = rb2[o]; v.y = rb2[o + 1]; cr2 = wmma_f32(a, v, cr2);
        v.x = rb3[o]; v.y = rb3[o + 1]; cr3 = wmma_f32(a, v, cr3);
    }
    const float invz = 1.f / (maxz[0] + EPSF);
    const float invr = 1.f / (maxr[0] + EPSF);
    float acc = 0.f;
    for (int nt = 0; nt < 4; ++nt) {
        const v8f* cz = (nt == 0) ? &cz0 : (nt == 1) ? &cz1 : (nt == 2) ? &cz2 : &cz3;
        const v8f* cr = (nt == 0) ? &cr0 : (nt == 1) ? &cr1 : (nt == 2) ? &cr2 : &cr3;
        for (int r = 0; r < 8; ++r) {
            int gi = ti * 16 + r + 8 * half;
            int gj = tj * 64 + nt * 16 + l;
            if (gi < gj) {
                float d2 = sqz[gi] + sqz[gj] - 2.f * (*cz)[r];
                float dz = (d2 > 0.f) ? sqrtf(d2) : 0.f;
                d2 = sqr[gi] + sqr[gj] - 2.f * (*cr)[r];
                float dr = (d2 > 0.f) ? sqrtf(d2) : 0.f;
                float zn = dz * invz, rn = dr * invr;
                float w = fmaxf(1.f - fmaxf(curv[gi], curv[gj]), 0.1f);
                float e = zn - rn;
                acc += e * e * w;
            }
        }
    }
    for (int o = 16; o > 0; o >>= 1) acc += __shfl_xor(acc, o);
    if (lane == 0 && acc != 0.f) atomicAdd(errs, acc);
}

// ---------------------------------------------------------------- finalize
__global__ void mmae_fin_kernel(const float* errs, float* out) {
    if (threadIdx.x == 0) {
        double np = (double)B_N * (double)(B_N - 1) / 2.0;
        out[0] = (float)((double)errs[0] / np);
    }
}

// ================================================================ launch
extern "C" void kernel_launch(void* const* d_in, const int* in_sizes, int n_in,
                              void* d_out, int out_size, void* d_ws, size_t ws_size,
                              hipStream_t stream) {
    const float* z   = (const float*)d_in[0];
    const float* ref = (const float*)d_in[1];
    const float* x   = (const float*)d_in[2];
    const int*   kp  = (const int*)d_in[3];
    float* out = (float*)d_out;

    float* ws   = (float*)d_ws;
    float* G    = ws;                              // B*B
    float* sqx  = G + (size_t)B_N * B_N;           // B
    float* sqz  = sqx + B_N;                       // B
    float* sqr  = sqz + B_N;                       // B
    float* curv = sqr + B_N;                       // B
    float* maxz = curv + B_N;                      // 1
    float* maxr = maxz + 1;                        // 1
    float* errs = maxr + 1;                        // 1
    int*   knn  = (int*)(errs + 1);                // B*KMAX

    mmae_init_kernel<<<1, 32, 0, stream>>>(maxz, maxr, errs, out);
    mmae_sqnorm_kernel<<<B_N, 256, 0, stream>>>(x,   D_X, sqx);
    mmae_sqnorm_kernel<<<B_N, 256, 0, stream>>>(z,   D_Z, sqz);
    mmae_sqnorm_kernel<<<B_N, 256, 0, stream>>>(ref, D_Z, sqr);

    dim3 strips(B_N / 64, B_N / 16);
    mmae_gemm_xxT_kernel<<<strips, 32, 0, stream>>>(x, G);
    mmae_topk_kernel<<<B_N, 256, 0, stream>>>(G, sqx, kp, knn);
    mmae_curv_kernel<<<B_N, 256, 0, stream>>>(x, knn, kp, curv);

    mmae_zr_max_kernel<<<strips, 32, 0, stream>>>(z, ref, sqz, sqr, maxz, maxr);
    mmae_err_kernel<<<strips, 32, 0, stream>>>(z, ref, sqz, sqr, curv, maxz, maxr, errs);
    mmae_fin_kernel<<<1, 32, 0, stream>>>(errs, out);
}